// FLD_45011257262249
// MI455X (gfx1250) — compile-verified
//
#include <hip/hip_runtime.h>

typedef __attribute__((ext_vector_type(2))) float v2f;
typedef __attribute__((ext_vector_type(8))) float v8f;

#define B_    64
#define T_    512
#define D_    96
#define TY_   512
#define H_    4
#define DK_   32
#define E_    128
#define P_    2
#define L_    64
#define HID_  128
#define HP_   8
#define TWO_D 192
#define AW_STRIDE 516   // 516 mod 64 == 4 -> 16 rows hit 16 distinct LDS banks

// workspace layout (float offsets)
#define WS_W2   0                                  // [E_][HP_]          = 1024
#define WS_C    1024                               // [HP_]              = 8
#define WS_WGT  2048                               // [B][HP_][T_]       = 262144
#define WS_XA   (2048 + B_*HP_*T_)                 // [B][P][H*2D=768]   = 98304
#define WS_CO   (WS_XA + B_*P_*768)                // [B][P][L]          = 8192

__device__ __forceinline__ v8f wmma4(v2f a, v2f b, v8f c) {
  // V_WMMA_F32_16X16X4_F32 : D = A(16x4,f32) * B(4x16,f32) + C(16x16,f32)
  return __builtin_amdgcn_wmma_f32_16x16x4_f32(
      /*neg_a=*/false, a, /*neg_b=*/false, b,
      /*c_mod=*/(short)0, c, /*reuse_a=*/false, /*reuse_b=*/false);
}

// ---------------------------------------------------------------- K0: fold q into W2
__global__ void k0_prep(const float* __restrict__ query, const float* __restrict__ q_w,
                        const float* __restrict__ q_b, const float* __restrict__ k_w,
                        const float* __restrict__ k_b, float* __restrict__ ws) {
  __shared__ float ql[P_][E_];
  int tid = threadIdx.x;
  {
    int p = tid >> 7, e = tid & 127;
    float acc = q_b[e];
    for (int i = 0; i < E_; ++i) acc += query[p * E_ + i] * q_w[i * E_ + e];
    ql[p][e] = acc;
  }
  __syncthreads();
  const float scale = 0.17677669529663687f;  // 1/sqrt(DK)
  for (int idx = tid; idx < E_ * HP_; idx += 256) {
    int e = idx >> 3, hp = idx & 7;
    int h = hp >> 1, p = hp & 1;
    float acc = 0.f;
    for (int d = 0; d < DK_; ++d) acc += k_w[e * E_ + h * DK_ + d] * ql[p][h * DK_ + d];
    ws[WS_W2 + e * HP_ + hp] = acc * scale;
  }
  if (tid < HP_) {
    int h = tid >> 1, p = tid & 1;
    float acc = 0.f;
    for (int d = 0; d < DK_; ++d) acc += k_b[h * DK_ + d] * ql[p][h * DK_ + d];
    ws[WS_C + tid] = acc * scale;
  }
}

// ------------------------------------------- K1: scores = emb@W2+c, then w = exp(s - rowmax)
__global__ void k1_scores(const float* __restrict__ timesteps, const float* __restrict__ te_w,
                          const float* __restrict__ te_b, float* __restrict__ ws) {
  __shared__ float tew[E_], teb[E_], w2l[E_ * HP_], cl[HP_], sl[HP_ * T_], mxl[HP_];
  int tid = threadIdx.x, b = blockIdx.x;
  if (tid < E_) { tew[tid] = te_w[tid]; teb[tid] = te_b[tid]; }
  for (int i = tid; i < E_ * HP_; i += 256) w2l[i] = ws[WS_W2 + i];
  if (tid < HP_) cl[tid] = ws[WS_C + tid];
  __syncthreads();
  for (int t = tid; t < T_; t += 256) {
    float ts = timesteps[b * T_ + t];
    float acc[HP_];
#pragma unroll
    for (int hp = 0; hp < HP_; ++hp) acc[hp] = cl[hp];
    for (int e = 0; e < E_; ++e) {
      float v = ts * tew[e] + teb[e];
      if ((e & 3) == 0) v = sinf(v);   // sin on indices where e % H == 0
#pragma unroll
      for (int hp = 0; hp < HP_; ++hp) acc[hp] += v * w2l[e * HP_ + hp];
    }
#pragma unroll
    for (int hp = 0; hp < HP_; ++hp) sl[hp * T_ + t] = acc[hp];
  }
  __syncthreads();
  int wv = tid >> 5, lane = tid & 31;
  {
    int hp = wv;  // 8 waves <-> 8 rows
    float m = -3.4e38f;
    for (int t = lane; t < T_; t += 32) m = fmaxf(m, sl[hp * T_ + t]);
#pragma unroll
    for (int off = 16; off > 0; off >>= 1) m = fmaxf(m, __shfl_xor(m, off, 32));
    if (lane == 0) mxl[hp] = m;
  }
  __syncthreads();
  for (int i = tid; i < HP_ * T_; i += 256) {
    int hp = i >> 9;
    ws[WS_WGT + b * HP_ * T_ + i] = expf(sl[i] - mxl[hp]);
  }
}

// ----------------------- K2: WMMA  num = w@(M*X), den = w@M over f<96 ; xa = num/den, 1.0 for f>=96
__global__ void k2_attn(const float* __restrict__ X, const float* __restrict__ M,
                        float* __restrict__ ws) {
  __shared__ float aw[16][AW_STRIDE];   // A matrix, rows 8..15 zero-padded
  __shared__ float nmS[HP_][D_];
  __shared__ float dnS[HP_][D_];
  int tid = threadIdx.x, b = blockIdx.x;
  int wv = tid >> 5, lane = tid & 31;
  int half = lane >> 4, nlane = lane & 15;
  const float* wgt = ws + WS_WGT + (size_t)b * HP_ * T_;
  const float* Xb = X + (size_t)b * T_ * D_;
  const float* Mb = M + (size_t)b * T_ * D_;
  // stage A into LDS: rows 0..7 = exp-weights, rows 8..15 = 0
  for (int i = tid; i < HP_ * T_; i += 256) aw[i >> 9][i & 511] = wgt[i];
  for (int i = tid; i < HP_ * T_; i += 256) aw[8 + (i >> 9)][i & 511] = 0.f;
  __syncthreads();
  for (int job = wv; job < 12; job += 8) {   // 6 num tiles + 6 den tiles, all f in [0,96)
    int isden = (job >= 6);                  // wave-uniform
    int n0 = (isden ? (job - 6) : job) * 16;
    int f = n0 + nlane;                      // always < 96
    v8f c = {0.f, 0.f, 0.f, 0.f, 0.f, 0.f, 0.f, 0.f};
    if (isden) {
      for (int k = 0; k < T_; k += 4) {
        int tk = k + 2 * half;               // lanes>=16 hold K = k+2, k+3
        v2f a, bb;
        a.x = aw[nlane][tk];
        a.y = aw[nlane][tk + 1];
        bb.x = Mb[(size_t)tk * D_ + f];
        bb.y = Mb[(size_t)(tk + 1) * D_ + f];
        c = wmma4(a, bb, c);
      }
    } else {
      for (int k = 0; k < T_; k += 4) {
        int tk = k + 2 * half;
        v2f a, bb;
        a.x = aw[nlane][tk];
        a.y = aw[nlane][tk + 1];
        float m0 = Mb[(size_t)tk * D_ + f];
        float m1 = Mb[(size_t)(tk + 1) * D_ + f];
        float x0 = Xb[(size_t)tk * D_ + f];
        float x1 = Xb[(size_t)(tk + 1) * D_ + f];
        bb.x = m0 * x0;
        bb.y = m1 * x1;
        c = wmma4(a, bb, c);
      }
    }
    if (lane < 16) {                         // lanes 0..15 carry rows g=0..7 (valid hp)
#pragma unroll
      for (int g = 0; g < 8; ++g) {
        if (isden) dnS[g][f] = c[g]; else nmS[g][f] = c[g];
      }
    }
  }
  __syncthreads();
  for (int i = tid; i < HP_ * TWO_D; i += 256) {
    int hp = i / TWO_D, f = i % TWO_D;
    int h = hp >> 1, p = hp & 1;             // xa.transpose(0,2,1,3) -> [b][p][h*2D+f]
    float val = (f < D_) ? (nmS[hp][f] / dnS[hp][f]) : 1.0f;  // mask-half attends to 1 exactly
    ws[WS_XA + b * (P_ * 768) + p * 768 + h * TWO_D + f] = val;
  }
}

// ----------------------------------------- K3: coeffs = xa @ ow + ob (small, VALU)
__global__ void k3_coeffs(const float* __restrict__ ow, const float* __restrict__ ob,
                          float* __restrict__ ws) {
  int i = blockIdx.x * 256 + threadIdx.x;  // B*P*L = 8192
  int b = i >> 7;
  int rem = i & 127;
  int p = rem >> 6, l = rem & 63;
  const float* xa = ws + WS_XA + b * (P_ * 768) + p * 768;
  float acc = ob[l];
  for (int j = 0; j < 768; ++j) acc += xa[j] * ow[j * L_ + l];
  ws[WS_CO + b * (P_ * L_) + p * L_ + l] = acc;
}

// ------------------- K4: fused decoder: x = c0 + y*c1 ; out = relu(x@w1+b1)@w2+b2
__global__ void k4_decoder(const float* __restrict__ y, const float* __restrict__ w1,
                           const float* __restrict__ b1, const float* __restrict__ w2,
                           const float* __restrict__ b2, const float* __restrict__ ws,
                           float* __restrict__ out) {
  __shared__ float xs[64][68];     // +4 pad: column-strided A reads hit distinct banks
  __shared__ float h1s[64][132];
  int tid = threadIdx.x;
  int b = blockIdx.x >> 3, chunk = blockIdx.x & 7;
  int ty0 = chunk * 64;
  int wv = tid >> 5, lane = tid & 31, half = lane >> 4, nlane = lane & 15;
  const float* c0 = ws + WS_CO + b * (P_ * L_);
  const float* c1 = c0 + L_;
  // stage 1: x tile in LDS
  for (int i = tid; i < 64 * 64; i += 256) {
    int r = i >> 6, l = i & 63;
    xs[r][l] = c0[l] + y[b * TY_ + ty0 + r] * c1[l];
  }
  __syncthreads();
  // stage 2: h1 = relu(x @ w1 + b1)  [64,64]@[64,128]
  for (int jj = 0; jj < 4; ++jj) {
    int job = wv * 4 + jj;         // 4 Mtiles x 8 Ntiles
    int m0 = (job >> 3) * 16, n0 = (job & 7) * 16;
    int n = n0 + nlane;
    float bias = b1[n];
    v8f c;
#pragma unroll
    for (int g = 0; g < 8; ++g) c[g] = bias;
    for (int k = 0; k < L_; k += 4) {
      int tk = k + 2 * half;
      v2f a, bb;
      a.x = xs[m0 + nlane][tk];
      a.y = xs[m0 + nlane][tk + 1];
      bb.x = w1[tk * HID_ + n];
      bb.y = w1[(tk + 1) * HID_ + n];
      c = wmma4(a, bb, c);
    }
#pragma unroll
    for (int g = 0; g < 8; ++g) h1s[m0 + g + half * 8][n] = fmaxf(c[g], 0.f);
  }
  __syncthreads();
  // stage 3: out = h1 @ w2 + b2  [64,128]@[128,96]
  for (int jj = 0; jj < 3; ++jj) {
    int job = wv * 3 + jj;         // 4 Mtiles x 6 Ntiles
    int m0 = (job / 6) * 16, n0 = (job % 6) * 16;
    int n = n0 + nlane;
    float bias = b2[n];
    v8f c;
#pragma unroll
    for (int g = 0; g < 8; ++g) c[g] = bias;
    for (int k = 0; k < HID_; k += 4) {
      int tk = k + 2 * half;
      v2f a, bb;
      a.x = h1s[m0 + nlane][tk];
      a.y = h1s[m0 + nlane][tk + 1];
      bb.x = w2[tk * D_ + n];
      bb.y = w2[(tk + 1) * D_ + n];
      c = wmma4(a, bb, c);
    }
#pragma unroll
    for (int g = 0; g < 8; ++g) {
      int row = ty0 + m0 + g + half * 8;
      out[((size_t)b * TY_ + row) * D_ + n] = c[g];
    }
  }
}

extern "C" void kernel_launch(void* const* d_in, const int* in_sizes, int n_in,
                              void* d_out, int out_size, void* d_ws, size_t ws_size,
                              hipStream_t stream) {
  (void)in_sizes; (void)n_in; (void)out_size; (void)ws_size;
  const float* timesteps = (const float*)d_in[0];
  const float* X     = (const float*)d_in[1];
  const float* M     = (const float*)d_in[2];
  const float* y     = (const float*)d_in[3];
  const float* te_w  = (const float*)d_in[4];
  const float* te_b  = (const float*)d_in[5];
  const float* query = (const float*)d_in[6];
  const float* q_w   = (const float*)d_in[7];
  const float* q_b   = (const float*)d_in[8];
  const float* k_w   = (const float*)d_in[9];
  const float* k_b   = (const float*)d_in[10];
  const float* ow    = (const float*)d_in[11];
  const float* ob    = (const float*)d_in[12];
  const float* w1    = (const float*)d_in[13];
  const float* b1    = (const float*)d_in[14];
  const float* w2    = (const float*)d_in[15];
  const float* b2    = (const float*)d_in[16];
  float* ws  = (float*)d_ws;
  float* out = (float*)d_out;

  hipLaunchKernelGGL(k0_prep,    dim3(1),      dim3(256), 0, stream, query, q_w, q_b, k_w, k_b, ws);
  hipLaunchKernelGGL(k1_scores,  dim3(B_),     dim3(256), 0, stream, timesteps, te_w, te_b, ws);
  hipLaunchKernelGGL(k2_attn,    dim3(B_),     dim3(256), 0, stream, X, M, ws);
  hipLaunchKernelGGL(k3_coeffs,  dim3(32),     dim3(256), 0, stream, ow, ob, ws);
  hipLaunchKernelGGL(k4_decoder, dim3(B_ * 8), dim3(256), 0, stream, y, w1, b1, w2, b2, ws, out);
}